// VectorQuantizerEMA_49615462203425
// MI455X (gfx1250) — compile-verified
//
#include <hip/hip_runtime.h>
#include <hip/hip_bf16.h>

typedef __attribute__((ext_vector_type(2))) float v2f;
typedef __attribute__((ext_vector_type(8))) float v8f;

#define K_CODES 4096
#define DIM     512
#define NROWS   16384   // 8 * 2048

// d_out layout (flat, return order): quantized_st (N*D), loss (1),
// encodings (N*K), encoding_indices (N)
#define QOFF     ((size_t)0)
#define LOSS_OFF ((size_t)NROWS * DIM)                    // 8388608
#define ENC_OFF  (LOSS_OFF + 1)                           // 8388609
#define IDX_OFF  (ENC_OFF + (size_t)NROWS * K_CODES)      // 75497473

// ---------------------------------------------------------------------------
// Kernel 0: w_sq per code (one wave per code) + zero counts / e-latent sum.
// grid = 512 blocks x 256 threads (8 waves/block -> 4096 codes)
// ---------------------------------------------------------------------------
__global__ __launch_bounds__(256)
void vq_prep(const float* __restrict__ emb, float* __restrict__ wsq,
             int* __restrict__ counts, float* __restrict__ esum) {
    const int t   = threadIdx.x;
    const int gid = blockIdx.x * 256 + t;
    if (gid < K_CODES) counts[gid] = 0;
    if (gid == 0) *esum = 0.0f;

    const int wave = t >> 5;
    const int lane = t & 31;
    const int code = blockIdx.x * 8 + wave;
    const float* row = emb + (size_t)code * DIM;
    float s = 0.0f;
    #pragma unroll
    for (int i = lane; i < DIM; i += 32) { float v = row[i]; s += v * v; }
    #pragma unroll
    for (int off = 16; off; off >>= 1) s += __shfl_down(s, off, 32);
    if (lane == 0) wsq[code] = s;
}

// ---------------------------------------------------------------------------
// Kernel 1: fused GEMM (fp32 WMMA) + argmin + quantized + counts + e-latent.
// grid = 1024 blocks (16 rows each) x 512 threads (16 waves).
// Each wave computes 16-row x 16-code tiles via v_wmma_f32_16x16x4_f32.
// Argmin keys (ordered-f32<<32 | code) are kept as PER-LANE running minima
// across all 16 code chunks; the cross-lane ds_bpermute reduction and the
// ds_min_u64 happen exactly once per wave at the end.
// ---------------------------------------------------------------------------
__global__ __launch_bounds__(512)
void vq_argmin(const float* __restrict__ x, const float* __restrict__ emb,
               const float* __restrict__ wsq_g, int* __restrict__ counts,
               float* __restrict__ esum, int* __restrict__ widx,
               float* __restrict__ out) {
    constexpr int XS = DIM + 4;  // pad: 512-stride would alias all 64 LDS banks
    __shared__ __align__(16) float xs[16 * XS];
    __shared__ float wsq_s[K_CODES];
    __shared__ unsigned long long minkey[16];
    __shared__ int   sidx[16];
    __shared__ float red[16];

    const int t    = threadIdx.x;
    const int wv   = t >> 5;
    const int lane = t & 31;
    const int half = lane >> 4;   // 0: K={0,1}, 1: K={2,3} of the wmma K-dim
    const int l16  = lane & 15;
    const int n0   = blockIdx.x * 16;

    // Stage 16 rows of x into LDS (float4 loads, 4 per thread).
    const float4* xg = reinterpret_cast<const float4*>(x + (size_t)n0 * DIM);
    #pragma unroll
    for (int i = 0; i < 4; ++i) {
        int f4   = i * 512 + t;           // 2048 float4s total
        int flat = f4 * 4;
        int row  = flat >> 9;
        int col  = flat & 511;
        float4 v = xg[f4];
        *reinterpret_cast<float4*>(&xs[row * XS + col]) = v;
    }
    for (int k = t; k < K_CODES; k += 512) wsq_s[k] = wsq_g[k];
    if (t < 16) minkey[t] = ~0ull;
    __syncthreads();

    unsigned long long runkey[8];
    #pragma unroll
    for (int j = 0; j < 8; ++j) runkey[j] = ~0ull;

    for (int cc = 0; cc < 16; ++cc) {            // 4096 / (16 waves * 16 codes)
        const int code_base = cc * 256 + wv * 16;
        const int mycode    = code_base + l16;   // this lane's code column
        const float* wrow   = emb + (size_t)mycode * DIM;
        const float* arow   = &xs[l16 * XS];     // this lane's x row (M = l16)
        v8f acc = {0.f, 0.f, 0.f, 0.f, 0.f, 0.f, 0.f, 0.f};
        #pragma unroll 4
        for (int ds = 0; ds < 128; ++ds) {       // D=512 in K=4 steps
            const int d0 = ds * 4 + 2 * half;
            v2f a; a.x = arow[d0]; a.y = arow[d0 + 1];   // A: 16x4 fp32
            v2f b; b.x = wrow[d0]; b.y = wrow[d0 + 1];   // B: 4x16 fp32
            acc = __builtin_amdgcn_wmma_f32_16x16x4_f32(
                false, a, false, b, (short)0, acc, false, false);
        }
        // C/D layout: VGPR j -> row j (lanes 0-15) / row j+8 (lanes 16-31),
        // N (code) = lane & 15 in both halves.
        const float wsqv = wsq_s[mycode];
        #pragma unroll
        for (int j = 0; j < 8; ++j) {
            float val = wsqv - 2.0f * acc[j];    // x_sq omitted: row-constant
            unsigned int u = __float_as_uint(val);
            u = (u & 0x80000000u) ? ~u : (u | 0x80000000u);  // order-preserving
            unsigned long long key =
                ((unsigned long long)u << 32) | (unsigned int)mycode;
            runkey[j] = (key < runkey[j]) ? key : runkey[j];
        }
    }

    // One cross-lane reduction per wave (within each 16-lane half), then one
    // ds_min_u64 per (row, wave).
    #pragma unroll
    for (int j = 0; j < 8; ++j) {
        unsigned long long key = runkey[j];
        #pragma unroll
        for (int off = 1; off < 16; off <<= 1) {
            unsigned long long o = __shfl_xor(key, off, 32);
            key = (o < key) ? o : key;
        }
        if (l16 == 0) atomicMin(&minkey[j + 8 * half], key);
    }
    __syncthreads();

    if (t < 16) {
        const int id = (int)(minkey[t] & 0xffffffffu);
        sidx[t] = id;
        widx[n0 + t] = id;
        out[IDX_OFF + n0 + t] = (float)id;
        atomicAdd(&counts[id], 1);
    }
    __syncthreads();

    // quantized_st == quantized numerically; also accumulate e-latent partial.
    // float4 path: thread handles rows i*4 + (t>>7), cols (t&127)*4 .. +3.
    float local = 0.0f;
    #pragma unroll
    for (int i = 0; i < 4; ++i) {
        int row  = i * 4 + (t >> 7);
        int col4 = (t & 127) * 4;
        const float4 q =
            *reinterpret_cast<const float4*>(emb + (size_t)sidx[row] * DIM + col4);
        const float4 xv =
            *reinterpret_cast<const float4*>(&xs[row * XS + col4]);
        *reinterpret_cast<float4*>(out + QOFF + (size_t)(n0 + row) * DIM + col4) = q;
        float dx = q.x - xv.x, dy = q.y - xv.y, dz = q.z - xv.z, dw = q.w - xv.w;
        local += dx * dx + dy * dy + dz * dz + dw * dw;
    }
    #pragma unroll
    for (int off = 16; off; off >>= 1) local += __shfl_down(local, off, 32);
    if (lane == 0) red[wv] = local;
    __syncthreads();
    if (t == 0) {
        float s = 0.0f;
        #pragma unroll
        for (int i = 0; i < 16; ++i) s += red[i];
        atomicAdd(esum, s);
    }
}

// ---------------------------------------------------------------------------
// Kernel 2: one-hot encodings — the dominant 268 MB store stream.
// Row base (ENC_OFF + n*4096) is odd, so peel 3 leading + 1 trailing scalar
// elements and use 16B-aligned float4 stores for k = 3 .. 4094.
// grid = 16384 blocks (one row each) x 256 threads.
// ---------------------------------------------------------------------------
__global__ __launch_bounds__(256)
void vq_onehot(const int* __restrict__ widx, float* __restrict__ enc) {
    const int n  = blockIdx.x;
    const int id = widx[n];
    float* row   = enc + (size_t)n * K_CODES;
    const int t  = threadIdx.x;
    if (t < 3) row[t] = (t == id) ? 1.0f : 0.0f;
    if (t == 3) row[K_CODES - 1] = ((K_CODES - 1) == id) ? 1.0f : 0.0f;
    float4* row4 = reinterpret_cast<float4*>(row + 3);   // 16B aligned
    #pragma unroll
    for (int q = t; q < 1023; q += 256) {                // covers k = 3..4094
        const int k = 3 + q * 4;
        float4 v;
        v.x = (k     == id) ? 1.0f : 0.0f;
        v.y = (k + 1 == id) ? 1.0f : 0.0f;
        v.z = (k + 2 == id) ? 1.0f : 0.0f;
        v.w = (k + 3 == id) ? 1.0f : 0.0f;
        row4[q] = v;
    }
}

// ---------------------------------------------------------------------------
// Kernel 3: finalize loss = 0.25*e_latent + 0.1*usage_loss (1 block).
// ---------------------------------------------------------------------------
__global__ __launch_bounds__(256)
void vq_loss(const int* __restrict__ counts, const float* __restrict__ esum,
             float* __restrict__ out_loss) {
    __shared__ float red[8];
    const int t = threadIdx.x;
    float s = 0.0f;
    for (int k = t; k < K_CODES; k += 256) {
        float u = (float)counts[k] * (1.0f / NROWS);
        float d = u - (1.0f / K_CODES);
        s += d * d;
    }
    #pragma unroll
    for (int off = 16; off; off >>= 1) s += __shfl_down(s, off, 32);
    if ((t & 31) == 0) red[t >> 5] = s;
    __syncthreads();
    if (t == 0) {
        float us = 0.0f;
        #pragma unroll
        for (int i = 0; i < 8; ++i) us += red[i];
        float usage_loss = us / K_CODES;
        float e_latent   = (*esum) / (float)((size_t)NROWS * DIM);
        *out_loss = 0.25f * e_latent + 0.1f * usage_loss;
    }
}

// ---------------------------------------------------------------------------
extern "C" void kernel_launch(void* const* d_in, const int* in_sizes, int n_in,
                              void* d_out, int out_size, void* d_ws, size_t ws_size,
                              hipStream_t stream) {
    (void)in_sizes; (void)n_in; (void)out_size; (void)ws_size;
    const float* x   = (const float*)d_in[0];   // (8,2048,512) fp32
    const float* emb = (const float*)d_in[1];   // (4096,512) fp32
    float* out = (float*)d_out;

    // workspace layout (floats): [0,4096) w_sq | [4096,8192) counts(int)
    //                            [8192] e-latent sum | [8704, 8704+N) idx(int)
    float* wsq    = (float*)d_ws;
    int*   counts = (int*)d_ws + 4096;
    float* esum   = (float*)d_ws + 8192;
    int*   widx   = (int*)d_ws + 8704;

    vq_prep  <<<512,   256, 0, stream>>>(emb, wsq, counts, esum);
    vq_argmin<<<1024,  512, 0, stream>>>(x, emb, wsq, counts, esum, widx, out);
    vq_onehot<<<16384, 256, 0, stream>>>(widx, out + ENC_OFF);
    vq_loss  <<<1,     256, 0, stream>>>(counts, esum, out + LOSS_OFF);
}